// TemporalRGCNEncoder_DGL_5763846111918
// MI455X (gfx1250) — compile-verified
//
#include <hip/hip_runtime.h>
#include <math.h>

// ---------------------------------------------------------------------------
// TemporalRGCNEncoder for MI455X (gfx1250, wave32).
//   N=50000 nodes, D=256, B=10 bases, R=240 relations, E=320000 edges.
// Strategy: node-side basis GEMMs via V_WMMA_F32_16X16X4_F32 (full fp32,
// 16x64 register-blocked per wave), then memory-bound edge gather + f32
// hardware atomics (roofline ~0.35ms at 23.3 TB/s).
// ---------------------------------------------------------------------------

#define NNODES 50000
#define NREL   240
#define DIM    256
#define NBASES 10
#define NEDGES 320000
#define DECAY  0.1f
#define LNEPS  1e-5f

typedef __attribute__((ext_vector_type(2))) float v2f;
typedef __attribute__((ext_vector_type(8))) float v8f;

// ---------------------------------------------------------------------------
// 1) max over edges of td = max(current_time - etimes, 0)
// ---------------------------------------------------------------------------
__global__ __launch_bounds__(256)
void tdmax_kernel(const float* __restrict__ etimes,
                  const float* __restrict__ ct,
                  unsigned* __restrict__ tdmax, int ne)
{
    float c = *ct;
    float m = 0.0f;
    for (int i = blockIdx.x * blockDim.x + threadIdx.x; i < ne;
         i += gridDim.x * blockDim.x)
        m = fmaxf(m, fmaxf(c - etimes[i], 0.0f));
    // wave32 reduction
    for (int o = 16; o > 0; o >>= 1)
        m = fmaxf(m, __shfl_xor(m, o, 32));
    if ((threadIdx.x & 31) == 0)
        atomicMax(tdmax, __float_as_uint(m));   // td >= 0 -> uint-bit monotone
}

// ---------------------------------------------------------------------------
// 2) enorm[e] = exp(-DECAY * td / (tdmax + 1e-8))
// ---------------------------------------------------------------------------
__global__ __launch_bounds__(256)
void enorm_kernel(const float* __restrict__ etimes,
                  const float* __restrict__ ct,
                  const unsigned* __restrict__ tdmax,
                  float* __restrict__ enorm, int ne)
{
    int i = blockIdx.x * blockDim.x + threadIdx.x;
    if (i >= ne) return;
    float mx = __uint_as_float(*tdmax);
    float td = fmaxf(*ct - etimes[i], 0.0f) / (mx + 1e-8f);
    enorm[i] = __expf(-DECAY * td);
}

// ---------------------------------------------------------------------------
// 3) Node-side GEMMs with fp32 WMMA, 16x64 strip per wave (4 accumulators).
//    blockIdx.z = w:  w <  loop_idx -> X[w] = H @ basis[b0+w]
//                     w == loop_idx -> agg  = H @ loop_w + bias
//    Block = 128 threads = 4 waves; wave `wv` covers cols [wv*64, wv*64+64).
//    Grid  = (N/16, 1, nw).
// ---------------------------------------------------------------------------
__global__ __launch_bounds__(128)
void rgcn_gemm_kernel(const float* __restrict__ H,
                      const float* __restrict__ basis,   // offset to b0
                      const float* __restrict__ loop_w,
                      const float* __restrict__ bias,
                      float* __restrict__ X,
                      float* __restrict__ agg,
                      int loop_idx)
{
    const int wv   = threadIdx.x >> 5;
    const int lane = threadIdx.x & 31;
    const int m0 = blockIdx.x * 16;
    const int n0 = wv * 64;
    const int w  = blockIdx.z;

    const bool isloop = (w == loop_idx);
    const float* __restrict__ W =
        isloop ? loop_w : (basis + (size_t)w * DIM * DIM);
    float* __restrict__ out =
        isloop ? agg : (X + (size_t)w * (size_t)NNODES * DIM);

    // A fragment (16x4 fp32): lanes 0-15 hold M=lane, K={k,k+1};
    //                         lanes 16-31 hold M=lane-16, K={k+2,k+3}.
    const int mrow = lane & 15;
    const int kh   = lane >> 4;          // 0: K+{0,1}, 1: K+{2,3}
    const float* __restrict__ arow = H + (size_t)(m0 + mrow) * DIM;

    v8f c[4] = {v8f{}, v8f{}, v8f{}, v8f{}};   // 4 x (16x16 fp32 accum)
#pragma unroll 2
    for (int k0 = 0; k0 < DIM; k0 += 4) {
        const int ka = k0 + 2 * kh;
        const v2f a = *(const v2f*)(arow + ka);          // one b64 load
        const float* __restrict__ w0 = W + (size_t)ka * DIM + n0 + mrow;
        const float* __restrict__ w1 = w0 + DIM;
#pragma unroll
        for (int j = 0; j < 4; ++j) {                    // reuse A x4
            v2f b;                                       // B 4x16 fragment
            b.x = w0[j * 16];
            b.y = w1[j * 16];
            c[j] = __builtin_amdgcn_wmma_f32_16x16x4_f32(
                       false, a, false, b, (short)0, c[j], false, false);
        }
    }

    // C layout: VGPR i -> row i (lanes 0-15) / row i+8 (lanes 16-31),
    //           col = lane&15.  Bias hoisted: one value per column strip.
#pragma unroll
    for (int j = 0; j < 4; ++j) {
        const int col = n0 + j * 16 + mrow;
        const float badd = isloop ? bias[col] : 0.0f;
        float* __restrict__ op = out + (size_t)(m0 + 8 * kh) * DIM + col;
#pragma unroll
        for (int i = 0; i < 8; ++i)
            op[(size_t)i * DIM] = c[j][i] + badd;
    }
}

// ---------------------------------------------------------------------------
// 4) Edge kernel: one wave per edge.
//    agg[dst] += enorm[e] * sum_b comp[etype, b0+b] * X[b][src]
//    Lane l covers d = {4l..4l+3} and {128+4l..128+4l+3} (coalesced b128).
// ---------------------------------------------------------------------------
__global__ __launch_bounds__(256)
void rgcn_edge_kernel(const float* __restrict__ X,
                      const float* __restrict__ comp,
                      const float* __restrict__ enorm,
                      const int* __restrict__ src,
                      const int* __restrict__ dst,
                      const int* __restrict__ ety,
                      float* __restrict__ agg,
                      int nb, int b0, int ne)
{
    const int wave = threadIdx.x >> 5;
    const int lane = threadIdx.x & 31;
    const int e = blockIdx.x * 8 + wave;
    if (e >= ne) return;

    const int s = src[e];
    const int d = dst[e];
    const int t = ety[e];
    const float en = enorm[e];

    const int d0 = lane * 4;
    const int d1 = 128 + lane * 4;

    float4 a0 = make_float4(0.f, 0.f, 0.f, 0.f);
    float4 a1 = make_float4(0.f, 0.f, 0.f, 0.f);

    for (int b = 0; b < nb; ++b) {
        const float cf = en * comp[(size_t)t * NBASES + b0 + b];
        const float* __restrict__ xr =
            X + ((size_t)b * NNODES + (size_t)s) * DIM;
        const float4 x0 = *(const float4*)(xr + d0);
        const float4 x1 = *(const float4*)(xr + d1);
        a0.x = fmaf(cf, x0.x, a0.x);  a0.y = fmaf(cf, x0.y, a0.y);
        a0.z = fmaf(cf, x0.z, a0.z);  a0.w = fmaf(cf, x0.w, a0.w);
        a1.x = fmaf(cf, x1.x, a1.x);  a1.y = fmaf(cf, x1.y, a1.y);
        a1.z = fmaf(cf, x1.z, a1.z);  a1.w = fmaf(cf, x1.w, a1.w);
    }

    float* __restrict__ ap = agg + (size_t)d * DIM;
    unsafeAtomicAdd(ap + d0 + 0, a0.x);   // global_atomic_add_f32
    unsafeAtomicAdd(ap + d0 + 1, a0.y);
    unsafeAtomicAdd(ap + d0 + 2, a0.z);
    unsafeAtomicAdd(ap + d0 + 3, a0.w);
    unsafeAtomicAdd(ap + d1 + 0, a1.x);
    unsafeAtomicAdd(ap + d1 + 1, a1.y);
    unsafeAtomicAdd(ap + d1 + 2, a1.z);
    unsafeAtomicAdd(ap + d1 + 3, a1.w);
}

// ---------------------------------------------------------------------------
// 5) h1 = relu(agg)  (vectorized elementwise)
// ---------------------------------------------------------------------------
__global__ __launch_bounds__(256)
void relu_copy_kernel(const float* __restrict__ in, float* __restrict__ out,
                      long n4)
{
    const float4* __restrict__ pi = (const float4*)in;
    float4* __restrict__ po = (float4*)out;
    for (long i = blockIdx.x * blockDim.x + threadIdx.x; i < n4;
         i += (long)gridDim.x * blockDim.x) {
        float4 v = pi[i];
        v.x = fmaxf(v.x, 0.f); v.y = fmaxf(v.y, 0.f);
        v.z = fmaxf(v.z, 0.f); v.w = fmaxf(v.w, 0.f);
        po[i] = v;
    }
}

// ---------------------------------------------------------------------------
// 6) out = LayerNorm(relu(agg) + emb) * g + b   (one wave per node)
// ---------------------------------------------------------------------------
__global__ __launch_bounds__(256)
void final_ln_kernel(const float* __restrict__ agg,
                     const float* __restrict__ emb,
                     const float* __restrict__ g,
                     const float* __restrict__ bb,
                     float* __restrict__ out, int nn)
{
    const int wave = threadIdx.x >> 5;
    const int lane = threadIdx.x & 31;
    const int n = blockIdx.x * 8 + wave;
    if (n >= nn) return;

    const float* __restrict__ ar = agg + (size_t)n * DIM;
    const float* __restrict__ er = emb + (size_t)n * DIM;

    float vals[8];
    float s = 0.f, s2 = 0.f;
#pragma unroll
    for (int j = 0; j < 8; ++j) {
        const int d = j * 32 + lane;
        const float v = fmaxf(ar[d], 0.f) + er[d];
        vals[j] = v; s += v; s2 += v * v;
    }
    for (int o = 16; o > 0; o >>= 1) {
        s  += __shfl_xor(s,  o, 32);
        s2 += __shfl_xor(s2, o, 32);
    }
    const float mu  = s * (1.0f / DIM);
    const float var = s2 * (1.0f / DIM) - mu * mu;
    const float rs  = rsqrtf(var + LNEPS);

    float* __restrict__ orow = out + (size_t)n * DIM;
#pragma unroll
    for (int j = 0; j < 8; ++j) {
        const int d = j * 32 + lane;
        orow[d] = (vals[j] - mu) * rs * g[d] + bb[d];
    }
}

// ---------------------------------------------------------------------------
// Host launcher
// ---------------------------------------------------------------------------
extern "C" void kernel_launch(void* const* d_in, const int* in_sizes, int n_in,
                              void* d_out, int out_size, void* d_ws,
                              size_t ws_size, hipStream_t stream)
{
    const float* emb    = (const float*)d_in[0];
    const float* basis1 = (const float*)d_in[1];
    const float* comp1  = (const float*)d_in[2];
    const float* lw1    = (const float*)d_in[3];
    const float* bias1  = (const float*)d_in[4];
    const float* basis2 = (const float*)d_in[5];
    const float* comp2  = (const float*)d_in[6];
    const float* lw2    = (const float*)d_in[7];
    const float* bias2  = (const float*)d_in[8];
    const float* lng    = (const float*)d_in[9];
    const float* lnb    = (const float*)d_in[10];
    const float* etimes = (const float*)d_in[11];
    const float* ct     = (const float*)d_in[12];
    const int*   src    = (const int*)d_in[13];
    const int*   dst    = (const int*)d_in[14];
    const int*   ety    = (const int*)d_in[15];
    float* out = (float*)d_out;

    // workspace layout (floats)
    float* ws = (float*)d_ws;
    const size_t ND = (size_t)NNODES * DIM;
    unsigned* tdmax = (unsigned*)ws;          // 1 uint (padded to 64 floats)
    float* enorm = ws + 64;                   // E
    float* agg   = enorm + NEDGES;            // N*D
    float* h1    = agg + ND;                  // N*D
    float* X     = h1 + ND;                   // bp * N*D

    const size_t fixed = 64 + (size_t)NEDGES + 2 * ND;
    const size_t availf = (ws_size / 4 > fixed) ? (ws_size / 4 - fixed) : 0;
    int bp = (int)(availf / ND);              // bases per pass, from ws_size
    if (bp > NBASES) bp = NBASES;
    if (bp < 1)      bp = 1;

    // time decay
    hipMemsetAsync(tdmax, 0, sizeof(unsigned), stream);
    tdmax_kernel<<<480, 256, 0, stream>>>(etimes, ct, tdmax, NEDGES);
    enorm_kernel<<<(NEDGES + 255) / 256, 256, 0, stream>>>(etimes, ct, tdmax,
                                                           enorm, NEDGES);

    const float* Hs[2]     = {emb, h1};
    const float* bases[2]  = {basis1, basis2};
    const float* comps[2]  = {comp1, comp2};
    const float* lws[2]    = {lw1, lw2};
    const float* biases[2] = {bias1, bias2};

    for (int L = 0; L < 2; ++L) {
        for (int b0 = 0; b0 < NBASES; b0 += bp) {
            const int nb = (NBASES - b0 < bp) ? (NBASES - b0) : bp;
            const int dl = (b0 == 0) ? 1 : 0;   // pass 0 also does loop_w+bias
            dim3 grid(NNODES / 16, 1, nb + dl);
            rgcn_gemm_kernel<<<grid, 128, 0, stream>>>(
                Hs[L], bases[L] + (size_t)b0 * DIM * DIM, lws[L], biases[L],
                X, agg, dl ? nb : -1);
            rgcn_edge_kernel<<<(NEDGES + 7) / 8, 256, 0, stream>>>(
                X, comps[L], enorm, src, dst, ety, agg, nb, b0, NEDGES);
        }
        if (L == 0)
            relu_copy_kernel<<<1250, 256, 0, stream>>>(agg, h1, (long)(ND / 4));
    }

    final_ln_kernel<<<(NNODES + 7) / 8, 256, 0, stream>>>(agg, emb, lng, lnb,
                                                          out, NNODES);
}